// MDRL_72121090834707
// MI455X (gfx1250) — compile-verified
//
#include <hip/hip_runtime.h>
#include <math.h>

#define NN   4096
#define FIN  512
#define HH1  512
#define HH2  256
#define NCL  8
#define CUT  128
#define AST  34    // A tile row stride (even -> aligned b64 k-pair fragments)
#define BPS  130   // packed B: kpair stride (64 n * 2 + 2 pad)
#define NEG_INF (-1e30f)

typedef __attribute__((ext_vector_type(2))) float v2f;
typedef __attribute__((ext_vector_type(8))) float v8f;
typedef long long ll;

// ---------------------------------------------------------------- WMMA helper
// CDNA5 fp32 tensor op: D(16x16) = A(16x4) * B(4x16) + C, wave32.
__device__ __forceinline__ v8f wmma_f32(v2f a, v2f b, v8f c) {
  return __builtin_amdgcn_wmma_f32_16x16x4_f32(false, a, false, b, (short)0, c,
                                               false, false);
}

// ---------------------------------------------------------------- reductions
__device__ __forceinline__ float block_sum(float v, float* red) {
  int t = threadIdx.x;
  red[t] = v; __syncthreads();
  for (int off = blockDim.x >> 1; off > 0; off >>= 1) {
    if (t < off) red[t] += red[t + off];
    __syncthreads();
  }
  float r = red[0]; __syncthreads();
  return r;
}
__device__ __forceinline__ float block_max(float v, float* red) {
  int t = threadIdx.x;
  red[t] = v; __syncthreads();
  for (int off = blockDim.x >> 1; off > 0; off >>= 1) {
    if (t < off) red[t] = fmaxf(red[t], red[t + off]);
    __syncthreads();
  }
  float r = red[0]; __syncthreads();
  return r;
}

__global__ void zero_kernel(float* p, ll n) {
  ll i = (ll)blockIdx.x * blockDim.x + threadIdx.x;
  ll st = (ll)gridDim.x * blockDim.x;
  for (; i < n; i += st) p[i] = 0.f;
}

// sum of a[r*ld + c], r<rows, c<cols  ->  per-block partials
__global__ void subsum_partial(const float* __restrict__ a, int rows, int cols,
                               int ld, float* __restrict__ partials) {
  __shared__ float red[256];
  ll n = (ll)rows * cols;
  ll i = (ll)blockIdx.x * blockDim.x + threadIdx.x;
  ll st = (ll)gridDim.x * blockDim.x;
  float s = 0.f;
  for (; i < n; i += st) {
    int r = (int)(i / cols), c = (int)(i % cols);
    s += a[(ll)r * ld + c];
  }
  s = block_sum(s, red);
  if (threadIdx.x == 0) partials[blockIdx.x] = s;
}

__global__ void final_sum(const float* __restrict__ partials, int n,
                          float* __restrict__ dst, float scale) {
  __shared__ float red[256];
  float s = 0.f;
  for (int i = threadIdx.x; i < n; i += blockDim.x) s += partials[i];
  s = block_sum(s, red);
  if (threadIdx.x == 0) *dst = s * scale;
}

// ------------------------------------------------ _dpn row normalization
__global__ void rownorm_kernel(const float* __restrict__ a, int cols, int ld,
                               const float* __restrict__ meanptr,
                               float* __restrict__ out) {
  __shared__ float red[256];
  int row = blockIdx.x, t = threadIdx.x;
  float mean = *meanptr;
  float v = (t < cols) ? (a[(ll)row * ld + t] - mean) : 0.f;
  float ss = block_sum(v * v, red);
  float nrm = fmaxf(sqrtf(ss), 1e-12f);
  if (t < cols) out[(ll)row * cols + t] = v / nrm;
}

// ---------------------------------------------------------------- WMMA GEMM
// C[M,Ncols] = rowscale(row) * (A[M,K] @ B[K,Ncols]) + bias(col), opt ReLU.
// 256 threads = 8 wave32; 128x64 block tile; each wave: 2x2 16x16 tiles.
// A tile row-major stride AST (b64 frags); B tile packed k-pairs (b64 frags).
__global__ __launch_bounds__(256)
void gemm_wmma(const float* __restrict__ A, const float* __restrict__ B,
               float* __restrict__ C, int M, int Ncols, int K,
               const float* __restrict__ bias,
               const float* __restrict__ rowscale, int relu) {
  __shared__ float As[128 * AST];
  __shared__ float Bs[16 * BPS];
  int t = threadIdx.x;
  int m0 = blockIdx.y * 128, n0 = blockIdx.x * 64;
  int wave = t >> 5, lane = t & 31;
  int mg = wave >> 1, ng = wave & 1;
  int lr = lane & 15, kb = (lane >> 4) * 2;
  v8f z = {0.f,0.f,0.f,0.f,0.f,0.f,0.f,0.f};
  v8f a00 = z, a01 = z, a10 = z, a11 = z;

  for (int k0 = 0; k0 < K; k0 += 32) {
    if (k0 + 32 < K)
      __builtin_prefetch(&A[(ll)(m0 + (t >> 1)) * K + k0 + 32], 0, 1);
    // A tile: 128x32 as 2048 b64 pairs
#pragma unroll
    for (int e = 0; e < 8; ++e) {
      int p = t + e * 256;
      int c2 = p & 15, r = p >> 4;
      *(v2f*)&As[r * AST + 2 * c2] =
          *(const v2f*)&A[(ll)(m0 + r) * K + k0 + 2 * c2];
    }
    // B tile: 32x64 scalars into packed k-pair layout
#pragma unroll
    for (int e = 0; e < 8; ++e) {
      int idx = t + e * 256;
      int r = idx >> 6, c = idx & 63;
      Bs[(r >> 1) * BPS + c * 2 + (r & 1)] = B[(ll)(k0 + r) * Ncols + (n0 + c)];
    }
    __syncthreads();
#pragma unroll
    for (int kk = 0; kk < 32; kk += 4) {
      v2f fa0 = *(const v2f*)&As[(mg * 32 + lr) * AST + kk + kb];
      v2f fa1 = *(const v2f*)&As[(mg * 32 + 16 + lr) * AST + kk + kb];
      int kp = (kk + kb) >> 1;
      v2f fb0 = *(const v2f*)&Bs[kp * BPS + (ng * 32 + lr) * 2];
      v2f fb1 = *(const v2f*)&Bs[kp * BPS + (ng * 32 + 16 + lr) * 2];
      a00 = wmma_f32(fa0, fb0, a00);
      a01 = wmma_f32(fa0, fb1, a01);
      a10 = wmma_f32(fa1, fb0, a10);
      a11 = wmma_f32(fa1, fb1, a11);
    }
    __syncthreads();
  }
#pragma unroll
  for (int v = 0; v < 8; ++v) {
    int r0 = m0 + mg * 32 + (lane >> 4) * 8 + v;
    int r1 = r0 + 16;
    int c0 = n0 + ng * 32 + lr, c1 = c0 + 16;
    float rs0 = rowscale ? rowscale[r0] : 1.f;
    float rs1 = rowscale ? rowscale[r1] : 1.f;
    float x00 = a00[v] * rs0, x01 = a01[v] * rs0;
    float x10 = a10[v] * rs1, x11 = a11[v] * rs1;
    if (bias) {
      float bc0 = bias[c0], bc1 = bias[c1];
      x00 += bc0; x01 += bc1; x10 += bc0; x11 += bc1;
    }
    if (relu) {
      x00 = fmaxf(x00, 0.f); x01 = fmaxf(x01, 0.f);
      x10 = fmaxf(x10, 0.f); x11 = fmaxf(x11, 0.f);
    }
    C[(ll)r0 * Ncols + c0] = x00;
    C[(ll)r0 * Ncols + c1] = x01;
    C[(ll)r1 * Ncols + c0] = x10;
    C[(ll)r1 * Ncols + c1] = x11;
  }
}

// ------------------------------------------ tadj = (X1n X1n^T) * (X2n X2n^T)
__global__ __launch_bounds__(256)
void dpn_pair_wmma(const float* __restrict__ X1, const float* __restrict__ X2,
                   float* __restrict__ out) {
  __shared__ float As1[128 * AST], As2[128 * AST];
  __shared__ float Bs1[16 * BPS], Bs2[16 * BPS];
  const int K = HH2;
  int t = threadIdx.x;
  int m0 = blockIdx.y * 128, n0 = blockIdx.x * 64;
  int wave = t >> 5, lane = t & 31;
  int mg = wave >> 1, ng = wave & 1;
  int lr = lane & 15, kb = (lane >> 4) * 2;
  v8f z = {0.f,0.f,0.f,0.f,0.f,0.f,0.f,0.f};
  v8f p100 = z, p101 = z, p110 = z, p111 = z;
  v8f p200 = z, p201 = z, p210 = z, p211 = z;

  for (int k0 = 0; k0 < K; k0 += 32) {
#pragma unroll
    for (int e = 0; e < 8; ++e) {
      int p = t + e * 256;
      int c2 = p & 15, r = p >> 4;
      *(v2f*)&As1[r * AST + 2 * c2] =
          *(const v2f*)&X1[(ll)(m0 + r) * K + k0 + 2 * c2];
      *(v2f*)&As2[r * AST + 2 * c2] =
          *(const v2f*)&X2[(ll)(m0 + r) * K + k0 + 2 * c2];
    }
    // transposed B tiles (rows n0..n0+63) as b64 loads into packed pairs
#pragma unroll
    for (int e = 0; e < 4; ++e) {
      int p = t + e * 256;
      int kp = p & 15, n = p >> 4;
      *(v2f*)&Bs1[kp * BPS + n * 2] =
          *(const v2f*)&X1[(ll)(n0 + n) * K + k0 + 2 * kp];
      *(v2f*)&Bs2[kp * BPS + n * 2] =
          *(const v2f*)&X2[(ll)(n0 + n) * K + k0 + 2 * kp];
    }
    __syncthreads();
#pragma unroll
    for (int kk = 0; kk < 32; kk += 4) {
      int kp = (kk + kb) >> 1;
      v2f a1m0 = *(const v2f*)&As1[(mg * 32 + lr) * AST + kk + kb];
      v2f a1m1 = *(const v2f*)&As1[(mg * 32 + 16 + lr) * AST + kk + kb];
      v2f b1n0 = *(const v2f*)&Bs1[kp * BPS + (ng * 32 + lr) * 2];
      v2f b1n1 = *(const v2f*)&Bs1[kp * BPS + (ng * 32 + 16 + lr) * 2];
      p100 = wmma_f32(a1m0, b1n0, p100);
      p101 = wmma_f32(a1m0, b1n1, p101);
      p110 = wmma_f32(a1m1, b1n0, p110);
      p111 = wmma_f32(a1m1, b1n1, p111);
      v2f a2m0 = *(const v2f*)&As2[(mg * 32 + lr) * AST + kk + kb];
      v2f a2m1 = *(const v2f*)&As2[(mg * 32 + 16 + lr) * AST + kk + kb];
      v2f b2n0 = *(const v2f*)&Bs2[kp * BPS + (ng * 32 + lr) * 2];
      v2f b2n1 = *(const v2f*)&Bs2[kp * BPS + (ng * 32 + 16 + lr) * 2];
      p200 = wmma_f32(a2m0, b2n0, p200);
      p201 = wmma_f32(a2m0, b2n1, p201);
      p210 = wmma_f32(a2m1, b2n0, p210);
      p211 = wmma_f32(a2m1, b2n1, p211);
    }
    __syncthreads();
  }
#pragma unroll
  for (int v = 0; v < 8; ++v) {
    int r0 = m0 + mg * 32 + (lane >> 4) * 8 + v;
    int r1 = r0 + 16;
    int c0 = n0 + ng * 32 + lr, c1 = c0 + 16;
    out[(ll)r0 * NN + c0] = p100[v] * p200[v];
    out[(ll)r0 * NN + c1] = p101[v] * p201[v];
    out[(ll)r1 * NN + c0] = p110[v] * p210[v];
    out[(ll)r1 * NN + c1] = p111[v] * p211[v];
  }
}

// ------------- fused edge_pre GEMM + stable BCE partial sums (up to 3 targets)
// logits l = (Xn Xn^T)[i,j]; sp = log1p(exp(-|l|)) + max(-l,0)
// Sa_t += (1-y)*l + sp ; Sb_t += y*sp   (pw applied later in finalize)
__global__ __launch_bounds__(256)
void edge_loss_wmma(const float* __restrict__ X, int K,
                    const float* __restrict__ Y1, const float* __restrict__ Y2,
                    const float* __restrict__ Y3, float* __restrict__ partials) {
  __shared__ float As[128 * AST];
  __shared__ float Bs[16 * BPS];
  __shared__ float red[256];
  int t = threadIdx.x;
  int m0 = blockIdx.y * 128, n0 = blockIdx.x * 64;
  int wave = t >> 5, lane = t & 31;
  int mg = wave >> 1, ng = wave & 1;
  int lr = lane & 15, kb = (lane >> 4) * 2;
  v8f z = {0.f,0.f,0.f,0.f,0.f,0.f,0.f,0.f};
  v8f a00 = z, a01 = z, a10 = z, a11 = z;

  for (int k0 = 0; k0 < K; k0 += 32) {
#pragma unroll
    for (int e = 0; e < 8; ++e) {
      int p = t + e * 256;
      int c2 = p & 15, r = p >> 4;
      *(v2f*)&As[r * AST + 2 * c2] =
          *(const v2f*)&X[(ll)(m0 + r) * K + k0 + 2 * c2];
    }
#pragma unroll
    for (int e = 0; e < 4; ++e) {
      int p = t + e * 256;
      int kp = p & 15, n = p >> 4;
      *(v2f*)&Bs[kp * BPS + n * 2] =
          *(const v2f*)&X[(ll)(n0 + n) * K + k0 + 2 * kp];
    }
    __syncthreads();
#pragma unroll
    for (int kk = 0; kk < 32; kk += 4) {
      int kp = (kk + kb) >> 1;
      v2f fa0 = *(const v2f*)&As[(mg * 32 + lr) * AST + kk + kb];
      v2f fa1 = *(const v2f*)&As[(mg * 32 + 16 + lr) * AST + kk + kb];
      v2f fb0 = *(const v2f*)&Bs[kp * BPS + (ng * 32 + lr) * 2];
      v2f fb1 = *(const v2f*)&Bs[kp * BPS + (ng * 32 + 16 + lr) * 2];
      a00 = wmma_f32(fa0, fb0, a00);
      a01 = wmma_f32(fa0, fb1, a01);
      a10 = wmma_f32(fa1, fb0, a10);
      a11 = wmma_f32(fa1, fb1, a11);
    }
    __syncthreads();
  }
  float s[6] = {0.f, 0.f, 0.f, 0.f, 0.f, 0.f};
#pragma unroll
  for (int i = 0; i < 2; ++i) {
#pragma unroll
    for (int j = 0; j < 2; ++j) {
#pragma unroll
      for (int v = 0; v < 8; ++v) {
        int row = m0 + mg * 32 + i * 16 + (lane >> 4) * 8 + v;
        int col = n0 + ng * 32 + j * 16 + lr;
        float l = i ? (j ? a11[v] : a10[v]) : (j ? a01[v] : a00[v]);
        float sp = log1pf(expf(-fabsf(l))) + fmaxf(-l, 0.f);
        ll yi = (ll)row * NN + col;
        { float y = Y1[yi]; s[0] += (1.f - y) * l + sp; s[1] += y * sp; }
        if (Y2) { float y = Y2[yi]; s[2] += (1.f - y) * l + sp; s[3] += y * sp; }
        if (Y3) { float y = Y3[yi]; s[4] += (1.f - y) * l + sp; s[5] += y * sp; }
      }
    }
  }
  int bid = blockIdx.y * gridDim.x + blockIdx.x;
  for (int q = 0; q < 6; ++q) {
    float r = block_sum(s[q], red);
    if (t == 0) partials[(ll)bid * 6 + q] = r;
  }
}

__global__ void edge_reduce(const float* __restrict__ partials, int nblocks,
                            float* __restrict__ out6) {
  __shared__ float red[256];
  for (int q = 0; q < 6; ++q) {
    float s = 0.f;
    for (int i = threadIdx.x; i < nblocks; i += blockDim.x)
      s += partials[(ll)i * 6 + q];
    s = block_sum(s, red);
    if (threadIdx.x == 0) out6[q] = s;
  }
}

// ---------------------------------------------- per-row top-k -> symmetric 0/1
__global__ __launch_bounds__(256)
void topk_mark(const float* __restrict__ tadj, float* __restrict__ tadjf,
               const int* __restrict__ kp) {
  int row = blockIdx.x, t = threadIdx.x;
  int k = *kp;
  float v[16];
#pragma unroll
  for (int j = 0; j < 16; ++j) v[j] = tadj[(ll)row * NN + t + j * 256];
  __shared__ float sv[256];
  __shared__ int si[256];
  for (int it = 0; it < k; ++it) {
    float best = NEG_INF; int bi = t;
#pragma unroll
    for (int j = 0; j < 16; ++j)
      if (v[j] > best) { best = v[j]; bi = t + j * 256; }
    sv[t] = best; si[t] = bi; __syncthreads();
    for (int off = 128; off > 0; off >>= 1) {
      if (t < off) {
        if (sv[t + off] > sv[t] ||
            (sv[t + off] == sv[t] && si[t + off] < si[t])) {
          sv[t] = sv[t + off]; si[t] = si[t + off];
        }
      }
      __syncthreads();
    }
    int col = si[0];
    if (t == (col & 255)) v[col >> 8] = NEG_INF;
    if (t == 0) {
      tadjf[(ll)row * NN + col] = 1.f;
      tadjf[(ll)col * NN + row] = 1.f;  // OR with transpose (writes of 1.0)
    }
    __syncthreads();
  }
}

// _tensor_normalize: r_inv[i] = 1/colsum_i (0 if empty)
__global__ void colsum_rinv(const float* __restrict__ tadjf,
                            float* __restrict__ rinv) {
  int c = blockIdx.x * blockDim.x + threadIdx.x;
  float s = 0.f;
  for (int r = 0; r < NN; ++r) s += (tadjf[(ll)r * NN + c] > 0.f) ? 1.f : 0.f;
  rinv[c] = (s > 0.f) ? (1.f / s) : 0.f;
}

// column sums of a normalized-half matrix (for simloss mean trick)
__global__ void half_colsum(const float* __restrict__ Xh, float* __restrict__ out) {
  int c = threadIdx.x;  // blockDim == CUT
  float s = 0.f;
  for (int r = 0; r < NN; ++r) s += Xh[(ll)r * CUT + c];
  out[c] = s;
}

__global__ void sim_dots(const float* __restrict__ hc, float* __restrict__ sc) {
  __shared__ float red[CUT];
  int t = threadIdx.x;
  const float* h1 = hc;
  const float* h2 = hc + CUT;
  const float* h3 = hc + 2 * CUT;
  float r12 = block_sum(h1[t] * h2[t], red);
  float r13 = block_sum(h1[t] * h3[t], red);
  float r32 = block_sum(h3[t] * h2[t], red);
  if (t == 0) { sc[40] = r12; sc[41] = r13; sc[42] = r32; }
}

// -------------------- attention fuse + log_softmax output + distill per row
__global__ __launch_bounds__(256)
void attn_out_distill(const float* __restrict__ e1, const float* __restrict__ e2,
                      const float* __restrict__ e3, const float* __restrict__ Wp1,
                      const float* __restrict__ bp1, const float* __restrict__ Wp2,
                      const float* __restrict__ Wm, const float* __restrict__ bm,
                      float* __restrict__ out, float* __restrict__ beta_out,
                      float* __restrict__ embf, float* __restrict__ rowd) {
  __shared__ float e[3][HH2];
  __shared__ float ef[HH2];
  __shared__ float th[3][16];
  __shared__ float wv[3], bet[3];
  __shared__ float lg[NCL], lstat[2];
  __shared__ float red[256];
  int row = blockIdx.x, t = threadIdx.x;
  e[0][t] = e1[(ll)row * HH2 + t];
  e[1][t] = e2[(ll)row * HH2 + t];
  e[2][t] = e3[(ll)row * HH2 + t];
  __syncthreads();
  if (t < 48) {
    int v = t >> 4, a = t & 15;
    float s = bp1[a];
    for (int kk = 0; kk < HH2; ++kk) s += e[v][kk] * Wp1[kk * 16 + a];
    th[v][a] = tanhf(s);
  }
  __syncthreads();
  if (t < 3) {
    float s = 0.f;
    for (int a = 0; a < 16; ++a) s += th[t][a] * Wp2[a];
    wv[t] = s;
  }
  __syncthreads();
  if (t == 0) {
    float m = fmaxf(wv[0], fmaxf(wv[1], wv[2]));
    float x0 = expf(wv[0] - m), x1 = expf(wv[1] - m), x2 = expf(wv[2] - m);
    float zz = x0 + x1 + x2;
    bet[0] = x0 / zz; bet[1] = x1 / zz; bet[2] = x2 / zz;
  }
  __syncthreads();
  if (t < 3) beta_out[(ll)row * 3 + t] = bet[t];
  float efv = bet[0] * e[0][t] + bet[1] * e[1][t] + bet[2] * e[2][t];
  ef[t] = efv;
  embf[(ll)row * HH2 + t] = efv;
  __syncthreads();
  if (t < NCL) {
    float s = bm[t];
    for (int j = 0; j < HH2; ++j) s += ef[j] * Wm[j * NCL + t];
    lg[t] = s;
  }
  __syncthreads();
  if (t == 0) {
    float m = lg[0];
    for (int c = 1; c < NCL; ++c) m = fmaxf(m, lg[c]);
    float zz = 0.f;
    for (int c = 0; c < NCL; ++c) zz += expf(lg[c] - m);
    lstat[0] = m; lstat[1] = logf(zz);
  }
  __syncthreads();
  if (t < NCL) out[(ll)row * NCL + t] = lg[t] - lstat[0] - lstat[1];
  // distill: p_t = softmax(emb_fused); sum over views of -(p_t * log_softmax(e_v))
  float m = block_max(efv, red);
  float zz = block_sum(expf(efv - m), red);
  float pt = expf(efv - m) / zz;
  float dacc = 0.f;
  for (int v = 0; v < 3; ++v) {
    float ev = e[v][t];
    float mv = block_max(ev, red);
    float zv = block_sum(expf(ev - mv), red);
    float lzv = logf(zv);
    dacc += block_sum(pt * (ev - mv - lzv), red);
  }
  if (t == 0) rowd[row] = -dacc;
}

// ---------------------------------------------------------------- finalize
__global__ void finalize(const float* __restrict__ sc, const float* __restrict__ es,
                         const float* __restrict__ srec_p,
                         const float* __restrict__ simv_p,
                         float* __restrict__ out3) {
  if (threadIdx.x != 0 || blockIdx.x != 0) return;
  const double n2 = (double)NN * (double)NN;
  auto lt = [=](float Sa, float Sb, float ysum) -> double {
    double norm = n2 / ((n2 - (double)ysum) * 2.0);
    double pw = (n2 - (double)ysum) / (double)ysum;
    return norm * ((double)Sa + (pw - 1.0) * (double)Sb) / n2;
  };
  float s = sc[0], f = sc[1], a2 = sc[2];
  double sl1 = lt(es[0], es[1], s);
  double dl1 = lt(es[2], es[3], f) + lt(es[4], es[5], a2);
  double sl2 = lt(es[6], es[7], f);
  double dl2 = lt(es[8], es[9], s) + lt(es[10], es[11], a2);
  double sl3 = lt(es[12], es[13], a2);
  double dl3 = lt(es[14], es[15], s) + lt(es[16], es[17], f);
  double rec = sl1 + dl1 + sl2 + dl2 + sl3 + dl3;
  double sim = (1.0 - (double)sc[40] / n2) + (1.0 - (double)sc[41] / n2) +
               (1.0 - (double)sc[42] / n2);
  out3[0] = (float)((double)(*srec_p) * rec + (double)(*simv_p) * sim);
  double spec = lt(es[18], es[19], s) + lt(es[24], es[25], f) +
                lt(es[30], es[31], a2);
  out3[1] = (float)spec;
  out3[2] = (float)((double)sc[43] / (3.0 * (double)NN));
}

// ================================================================ launcher
extern "C" void kernel_launch(void* const* d_in, const int* in_sizes, int n_in,
                              void* d_out, int out_size, void* d_ws,
                              size_t ws_size, hipStream_t stream) {
  (void)in_sizes; (void)n_in; (void)out_size; (void)ws_size;
  const float* x    = (const float*)d_in[0];
  const float* sadj = (const float*)d_in[1];
  const float* fadj = (const float*)d_in[2];
  const float* srec = (const float*)d_in[3];
  const float* simv = (const float*)d_in[4];
  const float* W1a  = (const float*)d_in[5];
  const float* b1a  = (const float*)d_in[6];
  const float* W2a  = (const float*)d_in[7];
  const float* b2a  = (const float*)d_in[8];
  const float* W1b  = (const float*)d_in[9];
  const float* b1b  = (const float*)d_in[10];
  const float* W2b  = (const float*)d_in[11];
  const float* b2b  = (const float*)d_in[12];
  const float* Wp1  = (const float*)d_in[13];
  const float* bp1  = (const float*)d_in[14];
  const float* Wp2  = (const float*)d_in[15];
  const float* Wm   = (const float*)d_in[16];
  const float* bm   = (const float*)d_in[17];
  const int*   kp   = (const int*)d_in[18];
  float* out = (float*)d_out;

  float* ws = (float*)d_ws;
  size_t off = 0;
  auto alloc = [&](size_t n) { float* p = ws + off; off += n; return p; };
  float* TADJ  = alloc((size_t)NN * NN);
  float* TADJF = alloc((size_t)NN * NN);
  float* T1A = alloc((size_t)NN * HH1);
  float* T1B = alloc((size_t)NN * HH1);
  float* HB  = alloc((size_t)NN * HH1);
  float* T2  = alloc((size_t)NN * HH2);
  float* E1  = alloc((size_t)NN * HH2);
  float* E2  = alloc((size_t)NN * HH2);
  float* E3  = alloc((size_t)NN * HH2);
  float* EF  = alloc((size_t)NN * HH2);
  float* NF1 = alloc((size_t)NN * HH2);
  float* NF2 = alloc((size_t)NN * HH2);
  float* NF3 = alloc((size_t)NN * HH2);
  float* NH1 = alloc((size_t)NN * CUT);
  float* NH2 = alloc((size_t)NN * CUT);
  float* NH3 = alloc((size_t)NN * CUT);
  float* RINV = alloc(NN);
  float* HC   = alloc(3 * CUT);
  float* ROWD = alloc(NN);
  float* SC   = alloc(64);
  float* ES   = alloc(48);
  float* PART = alloc(6 * 4096);

  dim3 blk(256);
  const int EBLK = (NN / 64) * (NN / 128);  // 2048 edge/dpn blocks
  auto sum_into = [&](const float* src, int rows, int cols, int ld, float* dst,
                      float scale) {
    subsum_partial<<<dim3(1024), blk, 0, stream>>>(src, rows, cols, ld, PART);
    final_sum<<<dim3(1), blk, 0, stream>>>(PART, 1024, dst, scale);
  };

  // ---- GCN 1: emb1 = sadj @ relu(sadj @ (x W1a) + b1a) W2a + b2a
  gemm_wmma<<<dim3(HH1/64, NN/128), blk, 0, stream>>>(x, W1a, T1A, NN, HH1, FIN, nullptr, nullptr, 0);
  gemm_wmma<<<dim3(HH1/64, NN/128), blk, 0, stream>>>(sadj, T1A, HB, NN, HH1, NN, b1a, nullptr, 1);
  gemm_wmma<<<dim3(HH2/64, NN/128), blk, 0, stream>>>(HB, W2a, T2, NN, HH2, HH1, nullptr, nullptr, 0);
  gemm_wmma<<<dim3(HH2/64, NN/128), blk, 0, stream>>>(sadj, T2, E1, NN, HH2, NN, b2a, nullptr, 0);
  // ---- GCN 2
  gemm_wmma<<<dim3(HH1/64, NN/128), blk, 0, stream>>>(x, W1b, T1B, NN, HH1, FIN, nullptr, nullptr, 0);
  gemm_wmma<<<dim3(HH1/64, NN/128), blk, 0, stream>>>(fadj, T1B, HB, NN, HH1, NN, b1b, nullptr, 1);
  gemm_wmma<<<dim3(HH2/64, NN/128), blk, 0, stream>>>(HB, W2b, T2, NN, HH2, HH1, nullptr, nullptr, 0);
  gemm_wmma<<<dim3(HH2/64, NN/128), blk, 0, stream>>>(fadj, T2, E2, NN, HH2, NN, b2b, nullptr, 0);

  // ---- _dpn normalization (full) of emb1/emb2, then tadj = sim1 * sim2
  sum_into(E1, NN, HH2, HH2, SC + 3, 1.f / ((float)NN * HH2));
  rownorm_kernel<<<dim3(NN), blk, 0, stream>>>(E1, HH2, HH2, SC + 3, NF1);
  sum_into(E2, NN, HH2, HH2, SC + 4, 1.f / ((float)NN * HH2));
  rownorm_kernel<<<dim3(NN), blk, 0, stream>>>(E2, HH2, HH2, SC + 4, NF2);
  dpn_pair_wmma<<<dim3(NN/64, NN/128), blk, 0, stream>>>(NF1, NF2, TADJ);

  // ---- top-k symmetric graph + tensor_normalize row scales
  zero_kernel<<<dim3(4096), blk, 0, stream>>>(TADJF, (ll)NN * NN);
  topk_mark<<<dim3(NN), blk, 0, stream>>>(TADJ, TADJF, kp);
  colsum_rinv<<<dim3(NN/256), blk, 0, stream>>>(TADJF, RINV);
  sum_into(sadj, NN, NN, NN, SC + 0, 1.f);
  sum_into(fadj, NN, NN, NN, SC + 1, 1.f);
  sum_into(TADJF, NN, NN, NN, SC + 2, 1.f);

  // ---- GCN 3 (tadj_n folded as row scale; reuses x @ W1b)
  gemm_wmma<<<dim3(HH1/64, NN/128), blk, 0, stream>>>(TADJF, T1B, HB, NN, HH1, NN, b1b, RINV, 1);
  gemm_wmma<<<dim3(HH2/64, NN/128), blk, 0, stream>>>(HB, W2b, T2, NN, HH2, HH1, nullptr, nullptr, 0);
  gemm_wmma<<<dim3(HH2/64, NN/128), blk, 0, stream>>>(TADJF, T2, E3, NN, HH2, NN, b2b, RINV, 0);

  // ---- remaining normalizations (emb3 full, halves of all three)
  sum_into(E3, NN, HH2, HH2, SC + 5, 1.f / ((float)NN * HH2));
  rownorm_kernel<<<dim3(NN), blk, 0, stream>>>(E3, HH2, HH2, SC + 5, NF3);
  sum_into(E1, NN, CUT, HH2, SC + 6, 1.f / ((float)NN * CUT));
  rownorm_kernel<<<dim3(NN), blk, 0, stream>>>(E1, CUT, HH2, SC + 6, NH1);
  sum_into(E2, NN, CUT, HH2, SC + 7, 1.f / ((float)NN * CUT));
  rownorm_kernel<<<dim3(NN), blk, 0, stream>>>(E2, CUT, HH2, SC + 7, NH2);
  sum_into(E3, NN, CUT, HH2, SC + 8, 1.f / ((float)NN * CUT));
  rownorm_kernel<<<dim3(NN), blk, 0, stream>>>(E3, CUT, HH2, SC + 8, NH3);

  // ---- fused edge_pre + BCE partial sums (shared: 3 targets; spec: 1 target)
  edge_loss_wmma<<<dim3(NN/64, NN/128), blk, 0, stream>>>(NH1, CUT, sadj, fadj, TADJF, PART);
  edge_reduce<<<dim3(1), blk, 0, stream>>>(PART, EBLK, ES + 0);
  edge_loss_wmma<<<dim3(NN/64, NN/128), blk, 0, stream>>>(NH2, CUT, fadj, sadj, TADJF, PART);
  edge_reduce<<<dim3(1), blk, 0, stream>>>(PART, EBLK, ES + 6);
  edge_loss_wmma<<<dim3(NN/64, NN/128), blk, 0, stream>>>(NH3, CUT, TADJF, sadj, fadj, PART);
  edge_reduce<<<dim3(1), blk, 0, stream>>>(PART, EBLK, ES + 12);
  edge_loss_wmma<<<dim3(NN/64, NN/128), blk, 0, stream>>>(NF1, HH2, sadj, nullptr, nullptr, PART);
  edge_reduce<<<dim3(1), blk, 0, stream>>>(PART, EBLK, ES + 18);
  edge_loss_wmma<<<dim3(NN/64, NN/128), blk, 0, stream>>>(NF2, HH2, fadj, nullptr, nullptr, PART);
  edge_reduce<<<dim3(1), blk, 0, stream>>>(PART, EBLK, ES + 24);
  edge_loss_wmma<<<dim3(NN/64, NN/128), blk, 0, stream>>>(NF3, HH2, TADJF, nullptr, nullptr, PART);
  edge_reduce<<<dim3(1), blk, 0, stream>>>(PART, EBLK, ES + 30);

  // ---- simloss via column-sum trick
  half_colsum<<<dim3(1), dim3(CUT), 0, stream>>>(NH1, HC + 0);
  half_colsum<<<dim3(1), dim3(CUT), 0, stream>>>(NH2, HC + CUT);
  half_colsum<<<dim3(1), dim3(CUT), 0, stream>>>(NH3, HC + 2 * CUT);
  sim_dots<<<dim3(1), dim3(CUT), 0, stream>>>(HC, SC);

  // ---- attention fuse, output log_softmax, beta, distill rows
  attn_out_distill<<<dim3(NN), blk, 0, stream>>>(E1, E2, E3, Wp1, bp1, Wp2, Wm, bm,
                                                 out, out + (size_t)NN * NCL, EF, ROWD);
  sum_into(ROWD, NN, 1, 1, SC + 43, 1.f);

  // ---- scalars: shared_loss, spec_loss, distill
  finalize<<<dim3(1), dim3(1), 0, stream>>>(SC, ES, srec, simv,
                                            out + (size_t)NN * NCL + (size_t)NN * 3);
}